// KVCache_17755394802340
// MI455X (gfx1250) — compile-verified
//
#include <hip/hip_runtime.h>
#include <hip/hip_bf16.h>
#include <stdint.h>

// Problem constants (from reference)
#define B_   2
#define H_   32
#define L_   8192
#define D_   128
#define S_   256
#define BH_  (B_ * H_)                       // 64
#define BHLD_ ((size_t)BH_ * L_ * D_)        // 67,108,864 elements per tensor

typedef float f32x4 __attribute__((ext_vector_type(4)));
typedef int   i32x4 __attribute__((ext_vector_type(4)));

// ---------------- CDNA5 async global->LDS copy (gfx1250) ----------------
// GLOBAL_LOAD_ASYNC_TO_LDS_B128: per-lane 16B  memory -> LDS, tracked by ASYNCcnt.
// VDST = per-lane LDS byte offset (low 32 bits of a shared-aperture pointer),
// VADDR = 64-bit global address, saddr disabled ("off").
__device__ __forceinline__ void async_load_b128(const void* gptr, const void* lds_ptr) {
  unsigned loff = (unsigned)(uintptr_t)lds_ptr;   // low 32 bits == LDS offset
  asm volatile("global_load_async_to_lds_b128 %0, %1, off"
               :: "v"(loff), "v"(gptr)
               : "memory");
}
#define WAIT_ASYNCCNT(n) asm volatile("s_wait_asynccnt " #n ::: "memory")

// ---------------- Kernel 1: dequantize q*scale+zp, streamed through LDS ----------------
// grid.x = BH_ * (L_/128) = 4096 blocks; block = 256 threads = 8 waves.
// Each wave: 16 rows, processed as 4 stages of 4 rows, double-buffered in LDS.
// LDS: 2 bufs * 8 waves * 4 rows * 128 dwords * 4B = 32 KB per block.
__global__ __launch_bounds__(256) void dequant_kernel(
    const int*   __restrict__ q,
    const float* __restrict__ sc,
    const float* __restrict__ zp,
    float*       __restrict__ out)
{
  __shared__ __align__(16) int lds[2][8][4][D_];

  const int tid  = threadIdx.x;
  const int wave = tid >> 5;
  const int lane = tid & 31;
  const int bh   = blockIdx.x >> 6;                 // /64 row-blocks per (b,h)
  const int rblk = blockIdx.x & 63;
  const int rowBase = (rblk << 7) + (wave << 4);    // 128 rows/block, 16 rows/wave
  const int dbase   = lane << 2;                    // 4 consecutive d per lane (16B)

  // per-(b,h,d) scale / zero-point, reused across all rows handled by this lane
  const f32x4 s4 = *(const f32x4*)(sc + bh * D_ + dbase);
  const f32x4 z4 = *(const f32x4*)(zp + bh * D_ + dbase);

  const int*  gbase = q   + ((size_t)bh * L_ + rowBase) * D_ + dbase;
  float*      obase = out + ((size_t)bh * L_ + rowBase) * D_ + dbase;

  // Prologue: stage 0 (4 rows) into buffer 0
#pragma unroll
  for (int r = 0; r < 4; ++r)
    async_load_b128(gbase + r * D_, &lds[0][wave][r][dbase]);

#pragma unroll
  for (int s = 0; s < 4; ++s) {
    const int cur = s & 1;
    if (s < 3) {
      // Prefetch stage s+1 into the other buffer, then wait until only those
      // 4 transfers remain outstanding (async loads complete in order).
      const int* g = gbase + (s + 1) * 4 * D_;
#pragma unroll
      for (int r = 0; r < 4; ++r)
        async_load_b128(g + r * D_, &lds[cur ^ 1][wave][r][dbase]);
      WAIT_ASYNCCNT(4);
    } else {
      WAIT_ASYNCCNT(0);
    }

    float* o = obase + s * 4 * D_;
#pragma unroll
    for (int r = 0; r < 4; ++r) {
      const i32x4 qi = *(const i32x4*)&lds[cur][wave][r][dbase];
      f32x4 f;
      f.x = fmaf((float)qi.x, s4.x, z4.x);
      f.y = fmaf((float)qi.y, s4.y, z4.y);
      f.z = fmaf((float)qi.z, s4.z, z4.z);
      f.w = fmaf((float)qi.w, s4.w, z4.w);
      // write-once 512 MiB stream: non-temporal, keep the 192MB L2 clean
      __builtin_nontemporal_store(f, (f32x4*)(o + r * D_));
    }
  }
}

// ---------------- Kernel 2: scatter fresh K/V rows at input_pos ----------------
// grid.x = BH_*S_/8 = 2048 blocks; block = 256 (8 rows/block, 32 lanes/row).
__global__ __launch_bounds__(256) void scatter_kernel(
    const float* __restrict__ val,     // [BH, S, D]
    const int*   __restrict__ pos,     // [S]
    float*       __restrict__ out)     // [BH, L, D]
{
  const int tid  = threadIdx.x;
  const int lane = tid & 31;
  const int row  = blockIdx.x * 8 + (tid >> 5);   // 0 .. BH*S-1
  const int bh   = row >> 8;                      // /S_
  const int s    = row & (S_ - 1);
  const int p    = pos[s];
  const int dbase = lane << 2;

  f32x4 v = *(const f32x4*)(val + (size_t)row * D_ + dbase);
  __builtin_nontemporal_store(v, (f32x4*)(out + ((size_t)bh * L_ + p) * D_ + dbase));
}

// ---------------- Kernel 3a: mask bool -> f32 ----------------
__global__ __launch_bounds__(256) void mask_convert_kernel(
    const unsigned char* __restrict__ m, float* __restrict__ out)
{
  const int i = blockIdx.x * 256 + threadIdx.x;   // BH_*L_ = 524288
  out[i] = m[i] ? 1.0f : 0.0f;
}

// ---------------- Kernel 3b: mask[.., input_pos] = True ----------------
__global__ __launch_bounds__(256) void mask_set_kernel(
    const int* __restrict__ pos, float* __restrict__ out)
{
  const int i  = blockIdx.x * 256 + threadIdx.x;  // BH_*S_ = 16384
  const int bh = i >> 8;
  const int s  = i & (S_ - 1);
  out[(size_t)bh * L_ + pos[s]] = 1.0f;
}

// ---------------- Launch ----------------
extern "C" void kernel_launch(void* const* d_in, const int* in_sizes, int n_in,
                              void* d_out, int out_size, void* d_ws, size_t ws_size,
                              hipStream_t stream) {
  (void)in_sizes; (void)n_in; (void)out_size; (void)d_ws; (void)ws_size;

  const int*   k_q  = (const int*)  d_in[0];
  const float* k_sc = (const float*)d_in[1];
  const float* k_zp = (const float*)d_in[2];
  const int*   v_q  = (const int*)  d_in[3];
  const float* v_sc = (const float*)d_in[4];
  const float* v_zp = (const float*)d_in[5];
  const float* k_val = (const float*)d_in[6];
  const float* v_val = (const float*)d_in[7];
  const int*   pos   = (const int*)  d_in[8];
  const unsigned char* mask_in = (const unsigned char*)d_in[9];

  float* out   = (float*)d_out;
  float* k_out = out;                       // stack([k, v]): k first
  float* v_out = out + BHLD_;
  float* m_out = out + 2 * BHLD_;

  const dim3 blk(256);

  // dequantize entire caches (stream-ordered, later kernels overwrite rows)
  dequant_kernel<<<BH_ * (L_ / 128), blk, 0, stream>>>(k_q, k_sc, k_zp, k_out);
  dequant_kernel<<<BH_ * (L_ / 128), blk, 0, stream>>>(v_q, v_sc, v_zp, v_out);

  // scatter new rows at input_pos
  scatter_kernel<<<(BH_ * S_) / 8, blk, 0, stream>>>(k_val, pos, k_out);
  scatter_kernel<<<(BH_ * S_) / 8, blk, 0, stream>>>(v_val, pos, v_out);

  // mask: convert then set True at input_pos
  mask_convert_kernel<<<(BH_ * L_) / 256, blk, 0, stream>>>(mask_in, m_out);
  mask_set_kernel<<<(BH_ * S_) / 256, blk, 0, stream>>>(pos, m_out);
}